// ATSSAssigner_32624571581008
// MI455X (gfx1250) — compile-verified
//
#include <hip/hip_runtime.h>
#include <hip/hip_bf16.h>
#include <math.h>

// ---------------- problem constants ----------------
#define BS        32
#define NMAX      32
#define BROWS     (BS * NMAX)        // 1024 gt rows
#define NA        8400               // anchors
#define TOPK_K    9
#define NLVL      3
#define NCAND     (TOPK_K * NLVL)    // 27
#define NCLS      80
#define BG_IDX    80

typedef __attribute__((ext_vector_type(2))) float v2f;
typedef __attribute__((ext_vector_type(8))) float v8f;

// ---------------- device helpers ----------------
__device__ __forceinline__ float iou2d_gt_anchor(const float* g, const float* a) {
    // matches _iou2d: eps=1e-6, union = max(a1+a2-ov, eps), no area clamp
    float area1 = (g[2] - g[0]) * (g[3] - g[1]);
    float area2 = (a[2] - a[0]) * (a[3] - a[1]);
    float ltx = fmaxf(g[0], a[0]);
    float lty = fmaxf(g[1], a[1]);
    float rbx = fminf(g[2], a[2]);
    float rby = fminf(g[3], a[3]);
    float w = fmaxf(rbx - ltx, 0.0f);
    float h = fmaxf(rby - lty, 0.0f);
    float ov = w * h;
    float un = fmaxf(area1 + area2 - ov, 1e-6f);
    return ov / un;
}

__device__ __forceinline__ float iou_batched_pair(const float* b1, const float* b2) {
    // matches _iou_batched: eps=1e-9, clamped areas, denom a1+a2-ov+eps
    float iw = fmaxf(fminf(b1[2], b2[2]) - fmaxf(b1[0], b2[0]), 0.0f);
    float ih = fmaxf(fminf(b1[3], b2[3]) - fmaxf(b1[1], b2[1]), 0.0f);
    float ov = iw * ih;
    float a1 = fmaxf(b1[2] - b1[0], 0.0f) * fmaxf(b1[3] - b1[1], 0.0f);
    float a2 = fmaxf(b2[2] - b2[0], 0.0f) * fmaxf(b2[3] - b2[1], 0.0f);
    return ov / (a1 + a2 - ov + 1e-9f);
}

// ---------------- K1: WMMA squared-distance matrix ----------------
// d2[r][a] = ||gt_center(r) - anchor_center(a)||^2
//          = [gx, gy, gn, 1] . [-2ax, -2ay, 1, an]   (K=4 GEMM)
// One wave computes one 16x16 tile with V_WMMA_F32_16X16X4_F32.
#define ROW_TILES 64            // 1024/16
#define COL_TILES 525          // 8400/16
#define WAVES_PER_BLK 8

__global__ void k_dist_wmma(const float* __restrict__ gt_bboxes,
                            const float* __restrict__ anc_bboxes,
                            float* __restrict__ d2) {
    const int lane = threadIdx.x & 31;
    const int wv   = threadIdx.x >> 5;
    const int tile = blockIdx.x * WAVES_PER_BLK + wv;   // 0 .. 33599 (exact)
    const int rt = tile / COL_TILES;
    const int ct = tile - rt * COL_TILES;

    const int m  = lane & 15;       // row within tile (A) / col within tile (B,D)
    const int hi = lane >> 4;       // 0 -> K{0,1}, 1 -> K{2,3}

    // ----- A operand: gt centers -----
    const int r = rt * 16 + m;                     // global gt row
    const float* gb = gt_bboxes + r * 4;
    float gx = 0.5f * (gb[0] + gb[2]);
    float gy = 0.5f * (gb[1] + gb[3]);
    float gn = gx * gx + gy * gy;
    v2f av;
    av.x = hi ? gn : gx;
    av.y = hi ? 1.0f : gy;

    // ----- B operand: anchor centers -----
    const int acol = ct * 16 + m;                  // global anchor col
    const float* ab = anc_bboxes + acol * 4;
    float ax = 0.5f * (ab[0] + ab[2]);
    float ay = 0.5f * (ab[1] + ab[3]);
    float an = ax * ax + ay * ay;
    v2f bv;
    bv.x = hi ? 1.0f : (-2.0f * ax);
    bv.y = hi ? an   : (-2.0f * ay);

    v8f c = {};
    // (neg_a, A, neg_b, B, c_mod, C, reuse_a, reuse_b)
    c = __builtin_amdgcn_wmma_f32_16x16x4_f32(false, av, false, bv,
                                              (short)0, c, false, false);

    // D layout: VGPR j -> M = j + 8*hi, N = lane&15
#pragma unroll
    for (int j = 0; j < 8; ++j) {
        int row = rt * 16 + j + 8 * hi;
        d2[row * NA + acol] = c[j];
    }
}

// ---------------- K2: overlaps = IoU(gt, anchor) ----------------
__global__ void k_overlaps(const float* __restrict__ gt_bboxes,
                           const float* __restrict__ anc_bboxes,
                           float* __restrict__ ov) {
    long long i = (long long)blockIdx.x * blockDim.x + threadIdx.x;
    if (i >= (long long)BROWS * NA) return;
    int bg = (int)(i / NA);
    int a  = (int)(i - (long long)bg * NA);
    ov[i] = iou2d_gt_anchor(gt_bboxes + bg * 4, anc_bboxes + a * 4);
}

// ---------------- K3: per-row per-level top-9 + threshold ----------------
// One wave32 per gt row. LDS copy of the level slice, 9 argmin passes.
__global__ __launch_bounds__(32)
void k_topk_thr(const float* __restrict__ d2,
                const float* __restrict__ ov,
                int* __restrict__ cand,
                float* __restrict__ thr) {
    __shared__ float sd[6400];
    __shared__ int   cidx[NCAND];

    const int bg   = blockIdx.x;
    const int lane = threadIdx.x;
    const int starts[NLVL] = {0, 6400, 8000};
    const int lens[NLVL]   = {6400, 1600, 400};

    int cc = 0;
    for (int lvl = 0; lvl < NLVL; ++lvl) {
        const int st = starts[lvl];
        const int L  = lens[lvl];
        for (int i = lane; i < L; i += 32) sd[i] = d2[(long long)bg * NA + st + i];
        __syncthreads();
        for (int p = 0; p < TOPK_K; ++p) {
            float bvv = 3.4e38f;
            int   bii = 0x7fffffff;
            for (int i = lane; i < L; i += 32) {
                float v = sd[i];
                if (v < bvv) { bvv = v; bii = i; }
            }
            // wave32 butterfly min-with-index (ties -> lowest index)
            for (int off = 16; off > 0; off >>= 1) {
                float vv = __shfl_xor(bvv, off, 32);
                int   ii = __shfl_xor(bii, off, 32);
                if (vv < bvv || (vv == bvv && ii < bii)) { bvv = vv; bii = ii; }
            }
            if (lane == 0) {
                cidx[cc + p] = st + bii;
                sd[bii] = 3.4e38f;    // exclude from later passes
            }
            __syncthreads();
        }
        cc += TOPK_K;
        __syncthreads();
    }

    // thr = mean + std(ddof=1) over the 27 candidate overlaps
    float x = 0.0f, x2 = 0.0f;
    int myidx = (lane < NCAND) ? cidx[lane] : 0;
    if (lane < NCAND) {
        float o = ov[(long long)bg * NA + myidx];
        x = o; x2 = o * o;
    }
    for (int off = 16; off > 0; off >>= 1) {
        x  += __shfl_xor(x,  off, 32);
        x2 += __shfl_xor(x2, off, 32);
    }
    if (lane == 0) {
        float mean = x / (float)NCAND;
        float var  = (x2 - (float)NCAND * mean * mean) / (float)(NCAND - 1);
        var = fmaxf(var, 0.0f);
        thr[bg] = mean + sqrtf(var);
    }
    if (lane < NCAND) cand[bg * NCAND + lane] = myidx;
}

// ---------------- K4: zero the mask buffer (reuses d2 storage) ----------------
__global__ void k_zero(float* __restrict__ p, long long n) {
    long long i = (long long)blockIdx.x * blockDim.x + threadIdx.x;
    if (i < n) p[i] = 0.0f;
}

// ---------------- K5: scatter positives among candidates ----------------
__global__ void k_scatter(const int* __restrict__ cand,
                          const float* __restrict__ thr,
                          const float* __restrict__ ov,
                          const float* __restrict__ gt_bboxes,
                          const float* __restrict__ anc_bboxes,
                          const float* __restrict__ mask_gt,
                          float* __restrict__ mask_pos) {
    int i = blockIdx.x * blockDim.x + threadIdx.x;
    if (i >= BROWS * NCAND) return;
    int bg = i / NCAND;
    int a  = cand[i];

    float o  = ov[(long long)bg * NA + a];
    float mg = mask_gt[bg];
    bool is_pos = o > thr[bg];

    const float* gb = gt_bboxes + bg * 4;
    const float* ab = anc_bboxes + a * 4;
    float acx = 0.5f * (ab[0] + ab[2]);
    float acy = 0.5f * (ab[1] + ab[3]);
    float mins = fminf(fminf(acx - gb[0], acy - gb[1]),
                       fminf(gb[2] - acx, gb[3] - acy));
    bool in_gts = mins > 1e-9f;

    float val = (is_pos && in_gts) ? mg : 0.0f;
    if (val != 0.0f) mask_pos[(long long)bg * NA + a] = val;
}

// ---------------- K6: per-anchor resolve + labels/bboxes/fg + score scratch ----------------
__global__ void k_resolve(const float* __restrict__ mask_pos,
                          const float* __restrict__ ov,
                          const float* __restrict__ gt_bboxes,
                          const int*   __restrict__ gt_labels,
                          const float* __restrict__ pd_bboxes,
                          float* __restrict__ out,          // full d_out base
                          int*   __restrict__ ws_lbl,
                          float* __restrict__ ws_sc) {
    int idx = blockIdx.x * blockDim.x + threadIdx.x;
    if (idx >= BS * NA) return;
    int b = idx / NA;
    int a = idx - b * NA;

    float mp[NMAX];
    float cnt = 0.0f;
    float bestov = -3.4e38f;
    int   ovarg  = 0;
#pragma unroll
    for (int g = 0; g < NMAX; ++g) {
        long long off = ((long long)(b * NMAX + g)) * NA + a;
        float m = mask_pos[off];
        mp[g] = m;
        cnt += m;
        float o = ov[off];
        if (o > bestov) { bestov = o; ovarg = g; }   // first max
    }
    bool multi = cnt > 1.0f;

    const float* pb = pd_bboxes + ((long long)b * NA + a) * 4;
    float fsum = 0.0f, bestmp = -1.0f, maxscore = 0.0f;
    int target = 0;
#pragma unroll
    for (int g = 0; g < NMAX; ++g) {
        float m = multi ? ((g == ovarg) ? 1.0f : 0.0f) : mp[g];
        fsum += m;
        if (m > bestmp) { bestmp = m; target = g; }   // argmax, first occurrence
        if (m > 0.0f) {
            const float* gb = gt_bboxes + (b * NMAX + g) * 4;
            float s = iou_batched_pair(gb, pb) * m;
            if (s > maxscore) maxscore = s;
        }
    }
    bool fg = fsum > 0.0f;
    int lbl = fg ? gt_labels[b * NMAX + target] : BG_IDX;

    // output regions (flat float, return-order concat)
    const long long OFF_LBL = 0;
    const long long OFF_BOX = (long long)BS * NA;                    //   268800
    const long long OFF_SCR = OFF_BOX + (long long)BS * NA * 4;      //  1344000
    const long long OFF_FG  = OFF_SCR + (long long)BS * NA * NCLS;   // 22848000

    out[OFF_LBL + idx] = (float)lbl;
    const float* tb = gt_bboxes + (b * NMAX + target) * 4;   // gathered unmasked, as in reference
#pragma unroll
    for (int c = 0; c < 4; ++c) out[OFF_BOX + (long long)idx * 4 + c] = tb[c];
    out[OFF_FG + idx] = fg ? 1.0f : 0.0f;

    ws_lbl[idx] = lbl;
    ws_sc[idx]  = maxscore;   // one_hot(BG_IDX,80) is all-zero, so bg rows vanish in K7
}

// ---------------- K7: coalesced 80-class score expansion ----------------
__global__ void k_scores(const int* __restrict__ ws_lbl,
                         const float* __restrict__ ws_sc,
                         float* __restrict__ out) {
    const long long OFF_SCR = (long long)BS * NA * 5;   // 268800 + 1075200
    long long i = (long long)blockIdx.x * blockDim.x + threadIdx.x;
    if (i >= (long long)BS * NA * NCLS) return;
    int ba = (int)(i / NCLS);
    int c  = (int)(i - (long long)ba * NCLS);
    float v = (ws_lbl[ba] == c) ? ws_sc[ba] : 0.0f;
    out[OFF_SCR + i] = v;
}

// ---------------- launcher ----------------
extern "C" void kernel_launch(void* const* d_in, const int* in_sizes, int n_in,
                              void* d_out, int out_size, void* d_ws, size_t ws_size,
                              hipStream_t stream) {
    const float* anc  = (const float*)d_in[0];   // (8400,4)
    const int*   glab = (const int*)  d_in[1];   // (32,32,1)
    const float* gtb  = (const float*)d_in[2];   // (32,32,4)
    const float* mgt  = (const float*)d_in[3];   // (32,32,1)
    const float* pdb  = (const float*)d_in[4];   // (32,8400,4)
    float* out = (float*)d_out;

    // workspace layout (bytes)
    char* ws = (char*)d_ws;
    const long long SZ_MAT = (long long)BROWS * NA * 4;   // 34,406,400
    float* d2_maskpos = (float*)(ws);                       // d2, later reused as mask_pos
    float* ov         = (float*)(ws + SZ_MAT);
    int*   cand       = (int*)  (ws + 2 * SZ_MAT);
    float* thr        = (float*)(ws + 2 * SZ_MAT + BROWS * NCAND * 4);
    int*   ws_lbl     = (int*)  (ws + 2 * SZ_MAT + BROWS * NCAND * 4 + BROWS * 4);
    float* ws_sc      = (float*)(ws + 2 * SZ_MAT + BROWS * NCAND * 4 + BROWS * 4
                                    + (long long)BS * NA * 4);

    // K1: WMMA distance^2 matrix  (64*525 = 33600 tiles, 8 waves/block)
    k_dist_wmma<<<(ROW_TILES * COL_TILES) / WAVES_PER_BLK, 32 * WAVES_PER_BLK, 0, stream>>>(
        gtb, anc, d2_maskpos);

    // K2: overlaps IoU matrix
    {
        long long n = (long long)BROWS * NA;
        k_overlaps<<<(int)((n + 255) / 256), 256, 0, stream>>>(gtb, anc, ov);
    }

    // K3: top-9 per level + threshold (one wave per gt row)
    k_topk_thr<<<BROWS, 32, 0, stream>>>(d2_maskpos, ov, cand, thr);

    // K4: zero mask buffer (reuse d2 storage)
    {
        long long n = (long long)BROWS * NA;
        k_zero<<<(int)((n + 255) / 256), 256, 0, stream>>>(d2_maskpos, n);
    }

    // K5: scatter candidate positives
    k_scatter<<<(BROWS * NCAND + 255) / 256, 256, 0, stream>>>(
        cand, thr, ov, gtb, anc, mgt, d2_maskpos);

    // K6: resolve per anchor, write labels/bboxes/fg + score scratch
    k_resolve<<<(BS * NA + 255) / 256, 256, 0, stream>>>(
        d2_maskpos, ov, gtb, glab, pdb, out, ws_lbl, ws_sc);

    // K7: score expansion (fully coalesced 86 MB write)
    {
        long long n = (long long)BS * NA * NCLS;
        k_scores<<<(int)((n + 255) / 256), 256, 0, stream>>>(ws_lbl, ws_sc, out);
    }
}